// NBFNet_6081673691198
// MI455X (gfx1250) — compile-verified
//
#include <hip/hip_runtime.h>
#include <hip/hip_bf16.h>
#include <math.h>

#define N_NODES   20000
#define NUM_EDGES 320000
#define BATCH     4
#define C         128
#define C2        256
#define ROWS      32       // rows (node,batch pairs) per workgroup in GEMM
#define US        260      // padded LDS stride for u (bank-conflict-free b64 reads)

typedef float v2f __attribute__((ext_vector_type(2)));
typedef float v8f __attribute__((ext_vector_type(8)));

// ---------------------------------------------------------------------------
// x = boundary : x[n,b,:] = (n == h_idx[b]) ? query : 0
// ---------------------------------------------------------------------------
__global__ __launch_bounds__(256)
void init_x_kernel(float* __restrict__ x, const int* __restrict__ h_idx,
                   const float* __restrict__ query) {
  size_t i = (size_t)blockIdx.x * blockDim.x + threadIdx.x;   // over N*B*C
  if (i >= (size_t)N_NODES * BATCH * C) return;
  int c = (int)(i & (C - 1));
  int b = (int)((i >> 7) & (BATCH - 1));
  int n = (int)(i >> 9);
  x[i] = (n == h_idx[b]) ? query[c] : 0.0f;
}

// ---------------------------------------------------------------------------
// CSR build, step 1: in-degree histogram (deg buffer pre-zeroed)
// ---------------------------------------------------------------------------
__global__ __launch_bounds__(256)
void degree_kernel(const int* __restrict__ dst, int* __restrict__ deg) {
  int e = blockIdx.x * blockDim.x + threadIdx.x;
  if (e >= NUM_EDGES) return;
  atomicAdd(&deg[dst[e]], 1);
}

// ---------------------------------------------------------------------------
// CSR build, step 2: exclusive scan of deg -> rowptr; also seed the slot
// cursor (cursor buffer aliases deg: read degree, then overwrite with start).
// Single block, 256 threads, chunk = 79 (256*79 >= 20000).
// ---------------------------------------------------------------------------
__global__ __launch_bounds__(256)
void scan_kernel(int* __restrict__ deg_cursor, int* __restrict__ rowptr) {
  __shared__ int part[256];
  const int tid = threadIdx.x;
  const int CH = 79;
  const int base = tid * CH;
  int s = 0;
  for (int i = 0; i < CH; ++i) {
    int n = base + i;
    if (n < N_NODES) s += deg_cursor[n];
  }
  part[tid] = s;
  __syncthreads();
  if (tid == 0) {
    int run = 0;
    for (int i = 0; i < 256; ++i) { int t = part[i]; part[i] = run; run += t; }
  }
  __syncthreads();
  int run = part[tid];
  for (int i = 0; i < CH; ++i) {
    int n = base + i;
    if (n < N_NODES) {
      int d = deg_cursor[n];
      rowptr[n] = run;
      deg_cursor[n] = run;     // becomes the fill cursor
      run += d;
    }
  }
  if (tid == 255) rowptr[N_NODES] = run;   // == NUM_EDGES
}

// ---------------------------------------------------------------------------
// CSR build, step 3: scatter (src, weight) into dst-sorted slots so the hot
// per-layer loop has no indirection.
// ---------------------------------------------------------------------------
__global__ __launch_bounds__(256)
void fill_kernel(const int* __restrict__ src, const int* __restrict__ dst,
                 const float* __restrict__ ew, int* __restrict__ cursor,
                 int* __restrict__ esrc, float* __restrict__ eww) {
  int e = blockIdx.x * blockDim.x + threadIdx.x;
  if (e >= NUM_EDGES) return;
  int slot = atomicAdd(&cursor[dst[e]], 1);
  esrc[slot] = src[e];
  eww[slot]  = ew[e];
}

// ---------------------------------------------------------------------------
// Destination-centric message aggregation (replaces atomic scatter):
//   agg[n,b,c] = sum_{e in CSR row n} rel[c] * w[e] * x[src[e],b,c]
// One block per node, 128 threads = (b, c4). Edge (src,w) staged in LDS once
// per 128-edge chunk; x rows are contiguous 2KB, L2-resident. Zero atomics.
// ---------------------------------------------------------------------------
__global__ __launch_bounds__(128)
void gather_kernel(const int* __restrict__ rowptr, const int* __restrict__ esrc,
                   const float* __restrict__ eww, const float* __restrict__ rel,
                   const float* __restrict__ x, float* __restrict__ agg) {
  __shared__ int   s_src[128];
  __shared__ float s_w[128];
  const int n   = blockIdx.x;
  const int tid = threadIdx.x;
  const int b   = tid >> 5;
  const int c4  = (tid & 31) << 2;
  const int beg = rowptr[n], end = rowptr[n + 1];
  const float4 rv = *(const float4*)(rel + c4);
  float4 acc = make_float4(0.f, 0.f, 0.f, 0.f);
  for (int base = beg; base < end; base += 128) {
    int cnt = min(128, end - base);
    if (tid < cnt) { s_src[tid] = esrc[base + tid]; s_w[tid] = eww[base + tid]; }
    __syncthreads();
    for (int j = 0; j < cnt; ++j) {
      const float4 xv = *(const float4*)(x + ((size_t)s_src[j] * BATCH + b) * C + c4);
      float w = s_w[j];
      acc.x += w * rv.x * xv.x;
      acc.y += w * rv.y * xv.y;
      acc.z += w * rv.z * xv.z;
      acc.w += w * rv.w * xv.w;
    }
    __syncthreads();
  }
  *(float4*)(agg + ((size_t)n * BATCH + b) * C + c4) = acc;
}

// ---------------------------------------------------------------------------
// Fused: h = [x | agg + rel*x + boundary] @ W^T + bias ; LayerNorm ; ReLU ;
//        x = result + x   (in-place, rows are workgroup-private)
// x-half of the LDS staging uses GLOBAL_LOAD_ASYNC_TO_LDS_B128 (ASYNCcnt),
// overlapped with the computed agg-half; WMMA = V_WMMA_F32_16X16X4_F32.
// ---------------------------------------------------------------------------
__global__ __launch_bounds__(256)
void gemm_ln_kernel(float* __restrict__ x, const float* __restrict__ agg,
                    const float* __restrict__ rel, const float* __restrict__ W,
                    const float* __restrict__ bias, const float* __restrict__ g,
                    const float* __restrict__ beta, const float* __restrict__ query,
                    const int* __restrict__ h_idx) {
  __shared__ __align__(16) float u[ROWS][US];        // [x | agg+rel*x+bdry]
  __shared__ __align__(16) float hbuf[ROWS][C + 4];  // pre-LN linear output
  __shared__ float red[ROWS][8][2];
  __shared__ float stats[ROWS][2];

  const int tid  = threadIdx.x;
  const int row0 = blockIdx.x * ROWS;

  // ---- async-copy the x half (k < C) straight into LDS ----
  for (int i = tid; i < ROWS * (C / 4); i += 256) {   // 1024 x 16B chunks
    int r = i >> 5;
    int k = (i & 31) << 2;
    unsigned long long ga = (unsigned long long)(x + (size_t)(row0 + r) * C + k);
    unsigned lds_off = (unsigned)(unsigned long long)&u[r][k];
    asm volatile("global_load_async_to_lds_b128 %0, %1, off"
                 :: "v"(lds_off), "v"(ga) : "memory");
  }

  // ---- compute the agg half (k >= C) while the copies are in flight ----
  for (int i = tid; i < ROWS * C; i += 256) {
    int r = i >> 7;
    int c = i & 127;
    int gr = row0 + r;
    int n = gr >> 2;
    int b = gr & 3;
    size_t off = (size_t)gr * C;
    float bd = (n == h_idx[b]) ? query[c] : 0.0f;
    u[r][C + c] = agg[off + c] + rel[c] * x[off + c] + bd;
  }

  asm volatile("s_wait_asynccnt 0x0" ::: "memory");
  __syncthreads();

  // ---- WMMA f32 16x16x4 accumulation ----
  const int lane = tid & 31;
  const int wave = tid >> 5;           // column tile 0..7
  const int m    = lane & 15;          // M (A) / N (B,C,D) index
  const int koff = (lane >> 4) << 1;   // lanes 0-15 -> K{0,1}; 16-31 -> K{2,3}
  const int colb = wave * 16;

  const float bias_v = bias[colb + m]; // D element depends only on N
  v8f c0, c1;
  for (int i = 0; i < 8; ++i) { c0[i] = bias_v; c1[i] = bias_v; }

  const float* wrow = W + (size_t)(colb + m) * C2;   // B[k][n] = W[colb+n][k]

  for (int k0 = 0; k0 < C2; k0 += 4) {
    v2f a0 = *(const v2f*)&u[m][k0 + koff];          // rows 0..15
    v2f a1 = *(const v2f*)&u[16 + m][k0 + koff];     // rows 16..31
    v2f bf = *(const v2f*)&wrow[k0 + koff];
    c0 = __builtin_amdgcn_wmma_f32_16x16x4_f32(false, a0, false, bf,
                                               (short)0, c0, false, false);
    c1 = __builtin_amdgcn_wmma_f32_16x16x4_f32(false, a1, false, bf,
                                               (short)0, c1, false, false);
  }

  // ---- spill D tiles to LDS for row-wise LayerNorm ----
  {
    int rbase = (lane >> 4) << 3;      // lanes 16-31 hold M = v+8
    for (int v = 0; v < 8; ++v) {
      hbuf[rbase + v][colb + m]      = c0[v];
      hbuf[16 + rbase + v][colb + m] = c1[v];
    }
  }
  __syncthreads();

  // ---- LayerNorm stats: 8 threads per row ----
  {
    int r = tid >> 3, part = tid & 7;
    float s = 0.f, s2 = 0.f;
    for (int c = part * 16; c < part * 16 + 16; ++c) {
      float v = hbuf[r][c];
      s += v; s2 += v * v;
    }
    red[r][part][0] = s; red[r][part][1] = s2;
  }
  __syncthreads();
  if (tid < ROWS) {
    float s = 0.f, s2 = 0.f;
    for (int p = 0; p < 8; ++p) { s += red[tid][p][0]; s2 += red[tid][p][1]; }
    float mu  = s * (1.0f / C);
    float var = s2 * (1.0f / C) - mu * mu;
    stats[tid][0] = mu;
    stats[tid][1] = rsqrtf(var + 1e-5f);
  }
  __syncthreads();

  // ---- normalize + ReLU + residual + writeback (old x == u[r][c]) ----
  for (int i = tid; i < ROWS * C; i += 256) {
    int r = i >> 7, c = i & 127;
    float hv = (hbuf[r][c] - stats[r][0]) * stats[r][1] * g[c] + beta[c];
    hv = hv > 0.f ? hv : 0.f;
    x[(size_t)(row0 + r) * C + c] = hv + u[r][c];
  }
}

// ---------------------------------------------------------------------------
// feat[b, :C] += x[t_idx[b], b, :] ; feat[b, C:] += query
// ---------------------------------------------------------------------------
__global__ __launch_bounds__(256)
void feat_kernel(float* __restrict__ feat, const float* __restrict__ x,
                 const int* __restrict__ t_idx, const float* __restrict__ query) {
  int i = blockIdx.x * blockDim.x + threadIdx.x;   // B*C = 512
  if (i >= BATCH * C) return;
  int b = i >> 7, c = i & 127;
  feat[b * C2 + c]     += x[((size_t)t_idx[b] * BATCH + b) * C + c];
  feat[b * C2 + C + c] += query[c];
}

// ---------------------------------------------------------------------------
// score = sigmoid( relu((feat/2) @ W1^T + b1) @ w2 + b2 ), single block
// ---------------------------------------------------------------------------
__global__ __launch_bounds__(256)
void mlp_kernel(const float* __restrict__ feat, const float* __restrict__ w1,
                const float* __restrict__ b1, const float* __restrict__ w2,
                const float* __restrict__ b2, float* __restrict__ out) {
  __shared__ float redv[BATCH][256];
  int j = threadIdx.x;
  float hj[BATCH];
  const float* wr = w1 + (size_t)j * C2;
  for (int b = 0; b < BATCH; ++b) {
    float acc = b1[j];
    for (int k = 0; k < C2; ++k) acc += 0.5f * feat[b * C2 + k] * wr[k];
    hj[b] = acc > 0.f ? acc : 0.f;
  }
  float w2j = w2[j];
  for (int b = 0; b < BATCH; ++b) redv[b][j] = hj[b] * w2j;
  __syncthreads();
  for (int off = 128; off > 0; off >>= 1) {
    if (j < off)
      for (int b = 0; b < BATCH; ++b) redv[b][j] += redv[b][j + off];
    __syncthreads();
  }
  if (j < BATCH) {
    float s = redv[j][0] + b2[0];
    out[j] = 1.0f / (1.0f + expf(-s));
  }
}

// ---------------------------------------------------------------------------
extern "C" void kernel_launch(void* const* d_in, const int* in_sizes, int n_in,
                              void* d_out, int out_size, void* d_ws, size_t ws_size,
                              hipStream_t stream) {
  const int*   eli   = (const int*)d_in[0];    // edge_label_index [2,B]
  const int*   ei    = (const int*)d_in[1];    // edge_index [2,E]
  const float* ew    = (const float*)d_in[2];  // [E]
  const float* rel_w = (const float*)d_in[3];  // [L,C]
  const float* lin_w = (const float*)d_in[4];  // [L,C,2C]
  const float* lin_b = (const float*)d_in[5];  // [L,C]
  const float* ln_g  = (const float*)d_in[6];  // [L,C]
  const float* ln_b  = (const float*)d_in[7];  // [L,C]
  const float* query = (const float*)d_in[8];  // [C]
  const float* w1    = (const float*)d_in[9];  // [2C,2C]
  const float* b1    = (const float*)d_in[10]; // [2C]
  const float* w2    = (const float*)d_in[11]; // [2C]
  const float* b2    = (const float*)d_in[12]; // [1]
  (void)in_sizes; (void)n_in; (void)out_size; (void)ws_size;

  const size_t nbc = (size_t)N_NODES * BATCH * C;
  float* x      = (float*)d_ws;
  float* agg    = x + nbc;
  float* feat   = agg + nbc;                    // B*2C floats
  int*   rowptr = (int*)(feat + BATCH * C2);    // N+1
  int*   cursor = rowptr + (N_NODES + 1);       // N (degree, then fill cursor)
  int*   esrc   = cursor + N_NODES;             // E (dst-sorted src)
  float* eww    = (float*)(esrc + NUM_EDGES);   // E (dst-sorted weight)
  float* out    = (float*)d_out;

  const int eblocks    = (NUM_EDGES + 255) / 256;
  const int initx_blocks = (int)((nbc + 255) / 256);
  const int gemm_blocks  = (N_NODES * BATCH) / ROWS;   // 2500, exact

  // ---- one-time CSR build (dst-sorted edges; kills per-layer atomics) ----
  hipMemsetAsync(cursor, 0, N_NODES * sizeof(int), stream);
  hipMemsetAsync(feat, 0, BATCH * C2 * sizeof(float), stream);
  degree_kernel<<<eblocks, 256, 0, stream>>>(ei + NUM_EDGES, cursor);
  scan_kernel<<<1, 256, 0, stream>>>(cursor, rowptr);
  fill_kernel<<<eblocks, 256, 0, stream>>>(ei, ei + NUM_EDGES, ew, cursor,
                                           esrc, eww);

  for (int dir = 0; dir < 2; ++dir) {
    const int* h_idx = eli + dir * BATCH;
    const int* t_idx = eli + (1 - dir) * BATCH;
    init_x_kernel<<<initx_blocks, 256, 0, stream>>>(x, h_idx, query);
    for (int l = 0; l < 3; ++l) {
      gather_kernel<<<N_NODES, 128, 0, stream>>>(rowptr, esrc, eww,
                                                 rel_w + l * C, x, agg);
      gemm_ln_kernel<<<gemm_blocks, 256, 0, stream>>>(
          x, agg, rel_w + l * C, lin_w + (size_t)l * C * C2,
          lin_b + l * C, ln_g + l * C, ln_b + l * C, query, h_idx);
    }
    feat_kernel<<<2, 256, 0, stream>>>(feat, x, t_idx, query);
  }
  mlp_kernel<<<1, 256, 0, stream>>>(feat, w1, b1, w2, b2, out);
}